// BidirectionalMambaBlocks_20392504722024
// MI455X (gfx1250) — compile-verified
//
#include <hip/hip_runtime.h>
#include <cstddef>

// ---------------------------------------------------------------------------
// Bidirectional Mamba block for MI455X (gfx1250, wave32).
// fp32 end-to-end; GEMMs use V_WMMA_F32_16X16X4_F32 with LDS double-buffered
// tiles (128x64 block tile, K-slab 16) so the inner loop is ds_load_b64 ->
// wmma with one barrier + one coarse loadcnt wait per slab.
// Selective scan is chunked (32 chunks x 128 steps) for 32x parallelism.
// ---------------------------------------------------------------------------

#define LSEQ   4096
#define BATCH  2
#define BL     8192      // BATCH * LSEQ
#define DMODEL 1024
#define DHALF  512
#define DINNER 1024
#define DSTATE 16
#define DTRANK 32
#define NCHUNK 32
#define CHUNK  128

#define KSLAB  16
#define APITCH 20        // 16 + 4 pad: conflict-free ds_load_b64 frag reads
#define BPITCH 20

typedef __attribute__((ext_vector_type(2))) float v2f;
typedef __attribute__((ext_vector_type(4))) float v4f;
typedef __attribute__((ext_vector_type(8))) float v8f;

__device__ __forceinline__ float silu_f(float x) { return x / (1.0f + expf(-x)); }

// ---------------------------------------------------------------------------
// LayerNorm over D=1024, then split: cols [0,512) -> xn_fwd (natural order),
// cols [512,1024) -> xn_bwd stored L-reversed (so the bwd Mamba runs forward).
// ---------------------------------------------------------------------------
__global__ void ln_split_kernel(const float* __restrict__ x,
                                const float* __restrict__ w,
                                const float* __restrict__ b,
                                float* __restrict__ xnf,
                                float* __restrict__ xnb) {
  const int row = blockIdx.x;                 // 0..BL-1
  const int tid = threadIdx.x;                // 256 threads
  const float* xr = x + (size_t)row * DMODEL;

  __shared__ float s1[256], s2[256];
  float sum = 0.f, sq = 0.f;
  for (int c = tid; c < DMODEL; c += 256) {
    float v = xr[c];
    sum += v; sq += v * v;
  }
  s1[tid] = sum; s2[tid] = sq;
  __syncthreads();
  for (int s = 128; s > 0; s >>= 1) {
    if (tid < s) { s1[tid] += s1[tid + s]; s2[tid] += s2[tid + s]; }
    __syncthreads();
  }
  const float mu  = s1[0] * (1.0f / DMODEL);
  const float var = s2[0] * (1.0f / DMODEL) - mu * mu;
  const float rs  = rsqrtf(var + 1e-5f);

  const int bat = row >> 12;           // row / LSEQ
  const int l   = row & (LSEQ - 1);
  for (int c = tid; c < DMODEL; c += 256) {
    float v = (xr[c] - mu) * rs * w[c] + b[c];
    if (c < DHALF) {
      xnf[(size_t)row * DHALF + c] = v;
    } else {
      int rrow = (bat << 12) + (LSEQ - 1 - l);
      xnb[(size_t)rrow * DHALF + (c - DHALF)] = v;
    }
  }
}

// ---------------------------------------------------------------------------
// Transpose W (Nrows x Kcols, row-major) -> Wt (Kcols x Nrows, row-major)
// so WMMA B tile loads are lane-coalesced b128s.
// ---------------------------------------------------------------------------
__global__ void transpose_kernel(const float* __restrict__ W,
                                 float* __restrict__ Wt,
                                 int Nrows, int Kcols) {
  int idx = blockIdx.x * blockDim.x + threadIdx.x;
  if (idx < Nrows * Kcols) {
    int n = idx / Kcols;
    int k = idx - n * Kcols;
    Wt[(size_t)k * Nrows + n] = W[idx];
  }
}

// ---------------------------------------------------------------------------
// C(M x N) = A(M x K, row stride lda) @ Bt(K x N, row-major)
// Block = 256 threads (8 waves), block tile 128(M) x 64(N), K-slab 16,
// LDS double-buffered.  Per wave: 16x64 (4 accumulators).
// Fragment layouts per CDNA5 ISA 7.12.2:
//   A 16x4 : a[j] = A[lane&15][k + 2*(lane>>4) + j]          (ds_load_b64)
//   B 4x16 : b[j] = Bt[k + 2*(lane>>4) + j][n0 + (lane&15)]  (ds_load_b64,
//            B staged transposed [n][k] in LDS, pitch 20 -> no bank conflicts)
//   C 16x16: acc[v] -> M = m0 + v + 8*(lane>>4), N = n0 + (lane&15)
// mode 0: plain store   mode 1: softplus(v + bias[col])
// mode 2: out-proj epilogue: residual add, optional L-reversal, col offset
// ---------------------------------------------------------------------------
__global__ void gemm_wmma_f32(const float* __restrict__ A, int lda,
                              const float* __restrict__ Bt,
                              float* __restrict__ C, int ldc,
                              int N, int K, int mode,
                              const float* __restrict__ bias,
                              const float* __restrict__ resid,
                              int reverse, int colOff) {
  __shared__ float sA[2][128 * APITCH];   // 10 KB each
  __shared__ float sB[2][64 * BPITCH];    //  5 KB each

  const int tid  = threadIdx.x;
  const int lane = tid & 31;
  const int wave = tid >> 5;
  const int half = lane >> 4;
  const int r    = lane & 15;
  const int mblk = blockIdx.y * 128;
  const int n0   = blockIdx.x * 64;

  // cooperative global-load mapping for one K-slab
  const int arow = tid >> 2;             // 0..63  (A rows arow and arow+64)
  const int acol = (tid & 3) * 4;        // 0,4,8,12
  const int brow = tid >> 4;             // 0..15  (B k-row)
  const int bcol = (tid & 15) * 4;       // 0..60  (B n-col, local)

  const float* gA0 = A + (size_t)(mblk + arow) * lda + acol;
  const float* gA1 = A + (size_t)(mblk + arow + 64) * lda + acol;
  const float* gB  = Bt + (size_t)brow * N + n0 + bcol;
  const size_t gBstep = (size_t)KSLAB * N;

  v8f acc[4] = { {}, {}, {}, {} };

  // ---- prologue: slab 0 ----
  v4f ra0 = *(const v4f*)gA0;
  v4f ra1 = *(const v4f*)gA1;
  v4f rb  = *(const v4f*)gB;
  sA[0][arow * APITCH + acol + 0] = ra0[0];
  sA[0][arow * APITCH + acol + 1] = ra0[1];
  sA[0][arow * APITCH + acol + 2] = ra0[2];
  sA[0][arow * APITCH + acol + 3] = ra0[3];
  sA[0][(arow + 64) * APITCH + acol + 0] = ra1[0];
  sA[0][(arow + 64) * APITCH + acol + 1] = ra1[1];
  sA[0][(arow + 64) * APITCH + acol + 2] = ra1[2];
  sA[0][(arow + 64) * APITCH + acol + 3] = ra1[3];
#pragma unroll
  for (int e = 0; e < 4; ++e)
    sB[0][(bcol + e) * BPITCH + brow] = rb[e];   // transpose to [n][k]
  __syncthreads();

  const int nslabs = K / KSLAB;
  int buf = 0;
  for (int s = 0; s < nslabs; ++s) {
    // issue next slab's global loads before compute (overlap with LDS math)
    v4f na0, na1, nb;
    const bool more = (s + 1) < nslabs;
    if (more) {
      na0 = *(const v4f*)(gA0 + (s + 1) * KSLAB);
      na1 = *(const v4f*)(gA1 + (s + 1) * KSLAB);
      nb  = *(const v4f*)(gB + (size_t)(s + 1) * gBstep);
      if (s + 2 < nslabs) {
        __builtin_prefetch(gA0 + (s + 2) * KSLAB, 0, 0);   // global_prefetch_b8
        __builtin_prefetch(gB + (size_t)(s + 2) * gBstep, 0, 0);
      }
    }

    // compute 16 WMMAs from LDS
    const float* As = &sA[buf][(wave * 16 + r) * APITCH + 2 * half];
    const float* Bs = &sB[buf][r * BPITCH + 2 * half];
#pragma unroll
    for (int kk = 0; kk < 4; ++kk) {
      v2f a = *(const v2f*)(As + kk * 4);
#pragma unroll
      for (int su = 0; su < 4; ++su) {
        v2f b = *(const v2f*)(Bs + su * 16 * BPITCH + kk * 4);
        acc[su] = __builtin_amdgcn_wmma_f32_16x16x4_f32(
            false, a, false, b, (short)0, acc[su], false, false);
      }
    }

    if (more) {
      const int nb_ = buf ^ 1;
      sA[nb_][arow * APITCH + acol + 0] = na0[0];
      sA[nb_][arow * APITCH + acol + 1] = na0[1];
      sA[nb_][arow * APITCH + acol + 2] = na0[2];
      sA[nb_][arow * APITCH + acol + 3] = na0[3];
      sA[nb_][(arow + 64) * APITCH + acol + 0] = na1[0];
      sA[nb_][(arow + 64) * APITCH + acol + 1] = na1[1];
      sA[nb_][(arow + 64) * APITCH + acol + 2] = na1[2];
      sA[nb_][(arow + 64) * APITCH + acol + 3] = na1[3];
#pragma unroll
      for (int e = 0; e < 4; ++e)
        sB[nb_][(bcol + e) * BPITCH + brow] = nb[e];
      __syncthreads();
      buf = nb_;
    }
  }

  // ---- epilogue ----
  const int m0 = mblk + wave * 16;
#pragma unroll
  for (int su = 0; su < 4; ++su) {
    const int col = n0 + su * 16 + r;
#pragma unroll
    for (int v = 0; v < 8; ++v) {
      const int row = m0 + v + 8 * half;
      float val = acc[su][v];
      if (mode == 1) {
        val += bias[col];
        val = (val > 20.0f) ? val : log1pf(expf(val));   // softplus
        C[(size_t)row * ldc + col] = val;
      } else if (mode == 2) {
        const int bat = row >> 12;
        const int l   = row & (LSEQ - 1);
        const int sl  = reverse ? (LSEQ - 1 - l) : l;
        const size_t oi = ((size_t)((bat << 12) + sl)) * DMODEL + colOff + col;
        C[oi] = val + resid[oi];
      } else {
        C[(size_t)row * ldc + col] = val;
      }
    }
  }
}

// ---------------------------------------------------------------------------
// Depthwise causal conv (K=4) + bias + SiLU on the first DINNER cols of xz.
// ---------------------------------------------------------------------------
__global__ void conv_silu_kernel(const float* __restrict__ xz,
                                 const float* __restrict__ cw,
                                 const float* __restrict__ cb,
                                 float* __restrict__ xc) {
  const int idx = blockIdx.x * blockDim.x + threadIdx.x;   // BL*DINNER threads
  const int d   = idx & (DINNER - 1);
  const int row = idx >> 10;
  const int l   = row & (LSEQ - 1);

  float acc = cb[d];
#pragma unroll
  for (int j = 0; j < 4; ++j) {
    const int lj = l - 3 + j;
    if (lj >= 0) acc += cw[d * 4 + j] * xz[(size_t)(row - 3 + j) * (2 * DINNER) + d];
  }
  xc[(size_t)row * DINNER + d] = silu_f(acc);
}

// ---------------------------------------------------------------------------
// Chunked selective scan, pass 1: per (b, chunk, d) compute
//   P[n]   = prod_i exp(dt_i * A[n])       (chunk decay)
//   hloc[n]= local scan end-state from h=0
// block: fixed (b, chunk), 256 channels -> B/C loads are wave-uniform (SMEM).
// ---------------------------------------------------------------------------
__global__ void scan_stats_kernel(const float* __restrict__ dt,
                                  const float* __restrict__ xc,
                                  const float* __restrict__ xdbl,
                                  const float* __restrict__ A_log,
                                  float* __restrict__ cP,
                                  float* __restrict__ cH) {
  const int blk   = blockIdx.x;               // b*128 + chunk*4 + dgrp
  const int dgrp  = blk & 3;
  const int chunk = (blk >> 2) & (NCHUNK - 1);
  const int bat   = blk >> 7;
  const int d     = dgrp * 256 + threadIdx.x;

  float An[DSTATE];
#pragma unroll
  for (int n = 0; n < DSTATE; ++n) An[n] = -expf(A_log[d * DSTATE + n]);

  float h[DSTATE], P[DSTATE];
#pragma unroll
  for (int n = 0; n < DSTATE; ++n) { h[n] = 0.f; P[n] = 1.f; }

  const size_t base = (size_t)(bat << 12) + chunk * CHUNK;
  for (int i = 0; i < CHUNK; ++i) {
    const size_t row = base + i;
    const float dtv = dt[row * DINNER + d];
    const float u   = xc[row * DINNER + d];
    const float du  = dtv * u;
    const float* Br = xdbl + row * 64 + DTRANK;    // B vector (uniform addr)
#pragma unroll
    for (int n = 0; n < DSTATE; ++n) {
      const float dA = expf(dtv * An[n]);
      h[n] = h[n] * dA + du * Br[n];
      P[n] *= dA;
    }
  }
  const size_t o = (((size_t)bat * NCHUNK + chunk) * DINNER + d) * DSTATE;
#pragma unroll
  for (int n = 0; n < DSTATE; ++n) { cP[o + n] = P[n]; cH[o + n] = h[n]; }
}

// ---------------------------------------------------------------------------
// Pass 2: sequential carry across the 32 chunks (cheap): h0[c] for each chunk.
// ---------------------------------------------------------------------------
__global__ void scan_carry_kernel(const float* __restrict__ cP,
                                  const float* __restrict__ cH,
                                  float* __restrict__ cH0) {
  const int idx = blockIdx.x * blockDim.x + threadIdx.x;   // BATCH*DINNER
  const int bat = idx >> 10;
  const int d   = idx & (DINNER - 1);

  float h0[DSTATE];
#pragma unroll
  for (int n = 0; n < DSTATE; ++n) h0[n] = 0.f;

  for (int c = 0; c < NCHUNK; ++c) {
    const size_t o = (((size_t)bat * NCHUNK + c) * DINNER + d) * DSTATE;
#pragma unroll
    for (int n = 0; n < DSTATE; ++n) cH0[o + n] = h0[n];
#pragma unroll
    for (int n = 0; n < DSTATE; ++n) h0[n] = cP[o + n] * h0[n] + cH[o + n];
  }
}

// ---------------------------------------------------------------------------
// Pass 3: replay each chunk from correct h0, emit fused gated output:
//   y = (C.h + D*u) * silu(z)
// ---------------------------------------------------------------------------
__global__ void scan_final_kernel(const float* __restrict__ dt,
                                  const float* __restrict__ xc,
                                  const float* __restrict__ xz,
                                  const float* __restrict__ xdbl,
                                  const float* __restrict__ A_log,
                                  const float* __restrict__ Dw,
                                  const float* __restrict__ cH0,
                                  float* __restrict__ y) {
  const int blk   = blockIdx.x;
  const int dgrp  = blk & 3;
  const int chunk = (blk >> 2) & (NCHUNK - 1);
  const int bat   = blk >> 7;
  const int d     = dgrp * 256 + threadIdx.x;

  float An[DSTATE];
#pragma unroll
  for (int n = 0; n < DSTATE; ++n) An[n] = -expf(A_log[d * DSTATE + n]);
  const float Dd = Dw[d];

  float h[DSTATE];
  {
    const size_t o = (((size_t)bat * NCHUNK + chunk) * DINNER + d) * DSTATE;
#pragma unroll
    for (int n = 0; n < DSTATE; ++n) h[n] = cH0[o + n];
  }

  const size_t base = (size_t)(bat << 12) + chunk * CHUNK;
  for (int i = 0; i < CHUNK; ++i) {
    const size_t row = base + i;
    const float dtv = dt[row * DINNER + d];
    const float u   = xc[row * DINNER + d];
    const float du  = dtv * u;
    const float* Br = xdbl + row * 64 + DTRANK;
    const float* Cr = xdbl + row * 64 + DTRANK + DSTATE;
    float ys = 0.f;
#pragma unroll
    for (int n = 0; n < DSTATE; ++n) {
      const float dA = expf(dtv * An[n]);
      h[n] = h[n] * dA + du * Br[n];
      ys += h[n] * Cr[n];
    }
    const float z = xz[row * (2 * DINNER) + DINNER + d];
    y[row * DINNER + d] = (ys + Dd * u) * silu_f(z);
  }
}

// ---------------------------------------------------------------------------
// Launch
// ---------------------------------------------------------------------------
extern "C" void kernel_launch(void* const* d_in, const int* in_sizes, int n_in,
                              void* d_out, int out_size, void* d_ws, size_t ws_size,
                              hipStream_t stream) {
  (void)in_sizes; (void)n_in; (void)out_size; (void)ws_size;

  const float* x      = (const float*)d_in[0];
  const float* norm_w = (const float*)d_in[1];
  const float* norm_b = (const float*)d_in[2];

  float* ws = (float*)d_ws;
  size_t o = 0;
  float* xn0   = ws + o; o += (size_t)BL * DHALF;
  float* xn1   = ws + o; o += (size_t)BL * DHALF;
  float* wt_in = ws + o; o += (size_t)DHALF * 2 * DINNER;       // 512 x 2048
  float* wt_xp = ws + o; o += (size_t)DINNER * 64;              // 1024 x 64
  float* wt_dt = ws + o; o += (size_t)DTRANK * DINNER;          // 32 x 1024
  float* wt_op = ws + o; o += (size_t)DINNER * DHALF;           // 1024 x 512
  float* xz    = ws + o; o += (size_t)BL * 2 * DINNER;
  float* xc    = ws + o; o += (size_t)BL * DINNER;
  float* xdbl  = ws + o; o += (size_t)BL * 64;
  float* dtb   = ws + o; o += (size_t)BL * DINNER;
  float* yb    = ws + o; o += (size_t)BL * DINNER;
  float* cP    = ws + o; o += (size_t)BATCH * NCHUNK * DINNER * DSTATE;
  float* cH    = ws + o; o += (size_t)BATCH * NCHUNK * DINNER * DSTATE;
  float* cH0   = ws + o; o += (size_t)BATCH * NCHUNK * DINNER * DSTATE;

  ln_split_kernel<<<BL, 256, 0, stream>>>(x, norm_w, norm_b, xn0, xn1);

  for (int dir = 0; dir < 2; ++dir) {
    const float* in_proj_w = (const float*)d_in[3 + 9 * dir];
    const float* conv_w    = (const float*)d_in[4 + 9 * dir];
    const float* conv_b    = (const float*)d_in[5 + 9 * dir];
    const float* x_proj_w  = (const float*)d_in[6 + 9 * dir];
    const float* dt_proj_w = (const float*)d_in[7 + 9 * dir];
    const float* dt_proj_b = (const float*)d_in[8 + 9 * dir];
    const float* A_log     = (const float*)d_in[9 + 9 * dir];
    const float* Dw        = (const float*)d_in[10 + 9 * dir];
    const float* out_projw = (const float*)d_in[11 + 9 * dir];
    const float* xn        = dir ? xn1 : xn0;

    transpose_kernel<<<(2 * DINNER * DHALF + 255) / 256, 256, 0, stream>>>(
        in_proj_w, wt_in, 2 * DINNER, DHALF);
    transpose_kernel<<<(64 * DINNER + 255) / 256, 256, 0, stream>>>(
        x_proj_w, wt_xp, 64, DINNER);
    transpose_kernel<<<(DINNER * DTRANK + 255) / 256, 256, 0, stream>>>(
        dt_proj_w, wt_dt, DINNER, DTRANK);
    transpose_kernel<<<(DHALF * DINNER + 255) / 256, 256, 0, stream>>>(
        out_projw, wt_op, DHALF, DINNER);

    // xz = xn @ in_proj_w^T  : (8192 x 512) @ (512 x 2048)
    gemm_wmma_f32<<<dim3((2 * DINNER) / 64, BL / 128), 256, 0, stream>>>(
        xn, DHALF, wt_in, xz, 2 * DINNER, 2 * DINNER, DHALF, 0,
        nullptr, nullptr, 0, 0);

    conv_silu_kernel<<<(BL * DINNER) / 256, 256, 0, stream>>>(xz, conv_w, conv_b, xc);

    // x_dbl = xc @ x_proj_w^T : (8192 x 1024) @ (1024 x 64)
    gemm_wmma_f32<<<dim3(64 / 64, BL / 128), 256, 0, stream>>>(
        xc, DINNER, wt_xp, xdbl, 64, 64, DINNER, 0, nullptr, nullptr, 0, 0);

    // dt = softplus(x_dbl[:, :32] @ dt_proj_w^T + b) : K=32, N=1024
    gemm_wmma_f32<<<dim3(DINNER / 64, BL / 128), 256, 0, stream>>>(
        xdbl, 64, wt_dt, dtb, DINNER, DINNER, DTRANK, 1,
        dt_proj_b, nullptr, 0, 0);

    scan_stats_kernel<<<BATCH * NCHUNK * 4, 256, 0, stream>>>(
        dtb, xc, xdbl, A_log, cP, cH);
    scan_carry_kernel<<<(BATCH * DINNER) / 256, 256, 0, stream>>>(cP, cH, cH0);
    scan_final_kernel<<<BATCH * NCHUNK * 4, 256, 0, stream>>>(
        dtb, xc, xz, xdbl, A_log, Dw, cH0, yb);

    // out = y @ out_proj_w^T (+ residual, reversal for bwd, concat offset)
    gemm_wmma_f32<<<dim3(DHALF / 64, BL / 128), 256, 0, stream>>>(
        yb, DINNER, wt_op, (float*)d_out, DMODEL, DHALF, DINNER, 2,
        nullptr, x, dir, dir * DHALF);
  }
}